// FocalCTCLoss_87600152969903
// MI455X (gfx1250) — compile-verified
//
#include <hip/hip_runtime.h>

typedef __attribute__((ext_vector_type(2))) float v2f;
typedef __attribute__((ext_vector_type(8))) float v8f;

#define TT 64
#define VV 96
#define LLAB 16
#define SS 33
#define NEGINF (-1e30f)
#define EPSF 1e-7f

__device__ __forceinline__ float lse2(float a, float b) {
    float m = fmaxf(a, b);
    return m + __logf(__expf(a - m) + __expf(b - m));
}
__device__ __forceinline__ float lse3(float a, float b, float c) {
    float m = fmaxf(a, fmaxf(b, c));
    return m + __logf(__expf(a - m) + __expf(b - m) + __expf(c - m));
}

// One wave32 per batch element. Low-LDS design (~8.7 KB) -> 32 WGs/WGP occupancy.
__global__ __launch_bounds__(32) void focal_ctc_kernel(const int* __restrict__ y_true,
                                                       const float* __restrict__ y_pred,
                                                       float* __restrict__ ws) {
    __shared__ float dlog[TT];            // log(rowsum + V*eps) per time step
    __shared__ float lp_lds[TT * SS];     // gathered per-state log-probs (8.4 KB)

    const int b = blockIdx.x;
    const int lane = threadIdx.x;
    const float* __restrict__ base = y_pred + (size_t)b * (TT * VV);

    // ---- Phase 0: prefetch the 24 KB tile (192 x 128B lines) into cache ----
#pragma unroll
    for (int i = 0; i < 6; ++i) {
        __builtin_prefetch(base + (i * 32 + lane) * 32, 0, 3);
    }

    // ---- Phase 1: row sums over V via f32 WMMA, A loaded directly from global ----
    // A layout (16x4 f32): lanes 0-15 hold {K=0,K=1}, lanes 16-31 hold {K=2,K=3},
    // row M = lane & 15. B = ones(4x16) -> every D column = rowsum[M].
    v2f bones; bones[0] = 1.0f; bones[1] = 1.0f;
    const int rowSel = lane & 15;
    const int klane  = (lane >> 4) << 1;
#pragma unroll
    for (int tb = 0; tb < 4; ++tb) {
        const int t0 = tb * 16;
        const float* rp = base + (t0 + rowSel) * VV + klane;   // 8B-aligned
        v8f c = {};
#pragma unroll
        for (int kb = 0; kb < 24; ++kb) {
            v2f a = *reinterpret_cast<const v2f*>(rp + kb * 4);
            c = __builtin_amdgcn_wmma_f32_16x16x4_f32(false, a, false, bones,
                                                      (short)0, c, false, false);
        }
        // C/D layout: VGPR r holds M=r (lanes 0-15) / M=8+r (lanes 16-31)
        if ((lane & 15) == 0) {
            int tb8 = t0 + ((lane >> 4) << 3);
#pragma unroll
            for (int r = 0; r < 8; ++r)
                dlog[tb8 + r] = __logf(c[r] + (float)VV * EPSF);
        }
    }
    __syncthreads();   // single-wave WG: S_NOP + compiler fence

    // ---- Phase 2: extended labels, skip mask, per-state log-prob gather ----
    const int s = lane;
    int ext_s = VV - 1;                   // blank
    bool skip = false;
    if (s & 1) {
        ext_s = y_true[b * LLAB + ((s - 1) >> 1)];
        if (s >= 3) skip = (ext_s != y_true[b * LLAB + ((s - 3) >> 1)]);
    }
    for (int t = 0; t < TT; ++t) {
        float val = __logf(base[t * VV + ext_s] + EPSF) - dlog[t];
        lp_lds[t * SS + s] = val;
        if (s == 0) lp_lds[t * SS + 32] = val;   // ext[32] == ext[0] == blank
    }
    __syncthreads();

    // ---- Phase 3: CTC forward scan; lane holds alpha[s], alpha[32] scalar ----
    float aL  = (lane < 2) ? lp_lds[lane] : NEGINF;
    float a32 = NEGINF;
    for (int t = 1; t < TT; ++t) {
        float lp   = lp_lds[t * SS + lane];
        float lp32 = __shfl(lp, 0);
        float a1   = aL;
        float up1  = __shfl_up(aL, 1);
        float up2  = __shfl_up(aL, 2);
        float a31v = __shfl(aL, 31);
        float a2 = (lane == 0) ? NEGINF : up1;
        float a3 = skip ? up2 : NEGINF;       // skip is false for lanes 0..2
        float n32 = lse2(a32, a31v) + lp32;   // state 32: blank, no skip path
        aL = lse3(a1, a2, a3) + lp;
        a32 = n32;
    }
    float a31f = __shfl(aL, 31);
    if (lane == 0) {
        float loglik = lse2(a32, a31f);
        float loss = -loglik;
        float p = __expf(loglik);
        float om = 1.0f - p;
        ws[b] = 2.0f * om * om * om * loss;   // ALPHA=2, GAMMA=3
    }
}

// Deterministic single-block tree reduction -> mean
__global__ __launch_bounds__(256) void focal_reduce_kernel(const float* __restrict__ ws,
                                                           float* __restrict__ out, int B) {
    __shared__ float sm[256];
    float sacc = 0.0f;
    for (int i = threadIdx.x; i < B; i += 256) sacc += ws[i];
    sm[threadIdx.x] = sacc;
    __syncthreads();
    for (int w = 128; w > 0; w >>= 1) {
        if ((int)threadIdx.x < w) sm[threadIdx.x] += sm[threadIdx.x + w];
        __syncthreads();
    }
    if (threadIdx.x == 0) out[0] = sm[0] / (float)B;
}

extern "C" void kernel_launch(void* const* d_in, const int* in_sizes, int n_in,
                              void* d_out, int out_size, void* d_ws, size_t ws_size,
                              hipStream_t stream) {
    const int*   y_true = (const int*)d_in[0];
    const float* y_pred = (const float*)d_in[1];
    const int B = in_sizes[1] / (TT * VV);
    float* ws = (float*)d_ws;

    focal_ctc_kernel<<<B, 32, 0, stream>>>(y_true, y_pred, ws);
    focal_reduce_kernel<<<1, 256, 0, stream>>>(ws, (float*)d_out, B);
}